// A3LALoss_65584150610448
// MI455X (gfx1250) — compile-verified
//
#include <hip/hip_runtime.h>
#include <hip/hip_bf16.h>
#include <math.h>

// Problem constants (from reference setup_inputs)
#define ENUM 3
#define BNUM 8192
#define CNUM 1000
#define SMARG 30.0f

typedef __attribute__((ext_vector_type(2))) float v2f;
typedef __attribute__((ext_vector_type(8))) float v8f;

// Workspace layout (float indices):
//   [0,3000)      sums[E][C]        energy segment sums
//   [3000,4000)   counts[C]
//   [4000,4006)   thetaSum[E][2]
//   [4006,4008)   thetaCnt[2]
//   [4008,4014)   meanReg[E][2]
//   [4014]        sumCls
//   [4016,4020)   int: L0, L1, R0, R1   (region bounds for masks 0/1)
//   [4020,7020)   logprior[E][C]  = log(prior^w + 1e-9)
#define WS_SUMS    0
#define WS_COUNTS  3000
#define WS_THSUM   4000
#define WS_THCNT   4006
#define WS_MEANREG 4008
#define WS_SUMCLS  4014
#define WS_REGION  4016
#define WS_LOGPRI  4020

__global__ void zero_kernel(float* ws, float* out0) {
    int i = blockIdx.x * blockDim.x + threadIdx.x;
    if (i < 4008) ws[i] = 0.0f;
    if (i == 0) *out0 = 0.0f;
}

// Single block: bitonic-sort cls counts in LDS, then serially replicate the
// numpy region-point walk (tiny: C=1000).
__global__ void region_kernel(const int* __restrict__ cls, float* ws) {
    __shared__ int s[1024];
    int tid = threadIdx.x;
    s[tid] = (tid < CNUM) ? cls[tid] : 0x7fffffff;
    __syncthreads();
    for (int k = 2; k <= 1024; k <<= 1) {
        for (int j = k >> 1; j > 0; j >>= 1) {
            int ixj = tid ^ j;
            if (ixj > tid) {
                int a = s[tid], b = s[ixj];
                bool up = ((tid & k) == 0);
                if ((a > b) == up) { s[tid] = b; s[ixj] = a; }
            }
            __syncthreads();
        }
    }
    if (tid == 0) {
        long long S = 0;
        for (int i = 0; i < CNUM; ++i) S += s[i];
        long long region_num = S / (ENUM - 1);
        int pts[8]; int np = 0;
        long long now = 0;
        for (int i = 0; i < CNUM; ++i) {
            now += s[i];
            if (now > region_num && np < 8) { pts[np++] = s[i]; now = 0; }
        }
        int maxCls = s[CNUM - 1];
        int L0, R0, L1, R1;
        if (np == 0) { L0 = 0; R0 = maxCls; L1 = 0; R1 = 0; }
        else {
            L0 = pts[np - 1]; R0 = maxCls;           // region 0 (head)
            if (np >= 2) { L1 = pts[np - 2]; R1 = pts[np - 1]; }
            else         { L1 = 0;           R1 = pts[np - 1]; }
        }
        int* iws = (int*)(ws + WS_REGION);
        iws[0] = L0; iws[1] = L1; iws[2] = R0; iws[3] = R1;
        ws[WS_SUMCLS] = (float)S;
    }
}

// One wave32 per sample: stable logsumexp over each expert's 64 features,
// atomic segment sums, theta region sums.
__global__ void persample_kernel(const float* __restrict__ feats,
                                 const float* __restrict__ logits,
                                 const int* __restrict__ target,
                                 const int* __restrict__ cls,
                                 float* ws) {
    int wid  = threadIdx.x >> 5;
    int lane = threadIdx.x & 31;
    int b = blockIdx.x * (blockDim.x >> 5) + wid;   // grid sized exactly

    const float* frow = feats + (size_t)b * (64 * ENUM);
    float energy[ENUM];
#pragma unroll
    for (int e = 0; e < ENUM; ++e) {
        float v0 = frow[e * 64 + lane];
        float v1 = frow[e * 64 + lane + 32];
        float m = fmaxf(v0, v1);
#pragma unroll
        for (int off = 16; off > 0; off >>= 1) m = fmaxf(m, __shfl_xor(m, off, 32));
        float sm = expf(v0 - m) + expf(v1 - m);
#pragma unroll
        for (int off = 16; off > 0; off >>= 1) sm += __shfl_xor(sm, off, 32);
        energy[e] = -(m + logf(sm));
    }
    if (lane == 0) {
        int t = target[b];
        float* sums   = ws + WS_SUMS;
        float* counts = ws + WS_COUNTS;
        float* thSum  = ws + WS_THSUM;
        float* thCnt  = ws + WS_THCNT;
        const int* iws = (const int*)(ws + WS_REGION);
        int L0 = iws[0], L1 = iws[1], R0 = iws[2], R1 = iws[3];
        atomicAdd(&counts[t], 1.0f);
        int tn = cls[t];
        int reg = (tn > L0 && tn <= R0) ? 0 : ((tn > L1 && tn <= R1) ? 1 : -1);
#pragma unroll
        for (int e = 0; e < ENUM; ++e) {
            atomicAdd(&sums[e * CNUM + t], energy[e]);
            float wyi = logits[((size_t)e * BNUM + b) * CNUM + t];
            float th  = acosf(wyi * (1.0f / SMARG));
            if (reg >= 0) atomicAdd(&thSum[e * 2 + reg], th);
        }
        if (reg >= 0) atomicAdd(&thCnt[reg], 1.0f);
    }
}

// One block per expert: means -> calib -> softmax -> w, weights out,
// logprior table, region means.
__global__ void stats_kernel(const int* __restrict__ cls, float* ws,
                             float* __restrict__ out) {
    __shared__ float red[1024];
    __shared__ float bcast;
    int e = blockIdx.x, tid = threadIdx.x;
    const float* sums   = ws + WS_SUMS;
    const float* counts = ws + WS_COUNTS;
    float* lp      = ws + WS_LOGPRI;
    float* meanReg = ws + WS_MEANREG;
    float sumCls = ws[WS_SUMCLS];

    float cnt = (tid < CNUM) ? counts[tid] : 0.0f;
    bool present = cnt > 0.0f;
    float mean = (tid < CNUM) ? sums[e * CNUM + tid] / fmaxf(cnt, 1.0f) : 0.0f;

    red[tid] = present ? mean : -INFINITY;
    __syncthreads();
    for (int s = 512; s > 0; s >>= 1) {
        if (tid < s) red[tid] = fmaxf(red[tid], red[tid + s]);
        __syncthreads();
    }
    if (tid == 0) bcast = red[0];
    __syncthreads();
    float maxE = bcast;

    float calib = present ? mean : maxE;    // absent classes -> max mean energy
    float ex = (tid < CNUM) ? expf(calib - maxE) : 0.0f;
    __syncthreads();
    red[tid] = ex;
    __syncthreads();
    for (int s = 512; s > 0; s >>= 1) {
        if (tid < s) red[tid] += red[tid + s];
        __syncthreads();
    }
    if (tid == 0) bcast = red[0];
    __syncthreads();
    float Z = bcast;

    if (tid < CNUM) {
        float w = 1.0f + ex / Z;
        out[1 + e * CNUM + tid] = w;                       // weights output
        float prior = (float)cls[tid] / sumCls;
        lp[e * CNUM + tid] = logf(powf(prior, w) + 1e-9f); // TAU==1 -> scale 1
    }
    if (tid < 2) {
        float* thSum = ws + WS_THSUM;
        float* thCnt = ws + WS_THCNT;
        meanReg[e * 2 + tid] = thSum[e * 2 + tid] / fmaxf(thCnt[tid], 1.0f);
    }
}

// ---------------------------------------------------------------------------
// Heavy pass: one wave per 16 logit rows.
// TDM (tensor_load_to_lds) double-buffers 16x64 f32 tiles global->LDS while
// the wave computes exp(adj) from the previous tile and accumulates per-row
// sums via V_WMMA_F32_16X16X4_F32 against an all-ones B matrix.
// ---------------------------------------------------------------------------
#define TILE_W 64
#define NTILE  15          /* 15*64 = 960 cols via TDM; tail 40 cols direct */

#if defined(__HIP_DEVICE_COMPILE__) && \
    __has_builtin(__builtin_amdgcn_tensor_load_to_lds) && \
    __has_builtin(__builtin_amdgcn_s_wait_tensorcnt)
#define USE_TDM 1
#else
#define USE_TDM 0
#endif

#if USE_TDM
typedef unsigned int u32x4 __attribute__((ext_vector_type(4)));
typedef int i32x8 __attribute__((ext_vector_type(8)));
typedef int i32x4 __attribute__((ext_vector_type(4)));

__device__ __forceinline__ unsigned lds_off(const void* p) {
    // generic -> LDS(addrspace 3) cast; AS3 pointer value == byte offset in LDS
    return (unsigned)(unsigned long long)
        (const __attribute__((address_space(3))) void*)p;
}

// Issue a 16-row x TILE_W-col f32 tile DMA: global (row-major, stride CNUM
// elements) -> LDS (dense row-major). D# per CDNA5 ISA ch.8.
__device__ __forceinline__ void tdm_issue_tile(const float* gtile,
                                               unsigned ldsByteOff) {
    unsigned long long ga = (unsigned long long)(uintptr_t)gtile;
    u32x4 g0;
    g0[0] = 1u;                                      // count=1, user mode
    g0[1] = ldsByteOff;                              // lds_addr (bytes)
    g0[2] = (unsigned)ga;                            // global_addr[31:0]
    g0[3] = (unsigned)((ga >> 32) & 0x01FFFFFFull)   // global_addr[56:32]
            | 0x80000000u;                           // type=2 ("image")
    i32x8 g1;
    g1[0] = 0x00020000;                              // data_size=2 -> 4 bytes
    g1[1] = (int)(((unsigned)TILE_W & 0xFFFFu) << 16);   // tensor_dim0 lo16
    g1[2] = (int)((16u & 0xFFFFu) << 16);            // dim0 hi=0 | tensor_dim1=16
    g1[3] = (int)(((unsigned)TILE_W & 0xFFFFu) << 16);   // dim1 hi=0 | tile_dim0
    g1[4] = 16;                                      // tile_dim1=16, tile_dim2=0
    g1[5] = CNUM;                                    // tensor_dim0_stride lo32
    g1[6] = 0;                                       // stride hi | dim1_stride lo
    g1[7] = 0;
    i32x4 z4 = {0, 0, 0, 0};                         // 2-D: groups 2/3 unused
    i32x8 z8 = {0, 0, 0, 0, 0, 0, 0, 0};             // unused extra group
    __builtin_amdgcn_tensor_load_to_lds(g0, g1, z4, z4, z8, 0);
    __asm__ __volatile__("" ::: "memory");
}
#endif

__global__ void __launch_bounds__(128)
loss_kernel(const float* __restrict__ logits,
            const int* __restrict__ target,
            const int* __restrict__ cls,
            const float* ws, float* __restrict__ out) {
    __shared__ __align__(16) float s_lp[CNUM];
#if USE_TDM
    __shared__ __align__(16) float s_tile[4][2][16 * TILE_W];  // 32 KB
#endif
    const float* lp      = ws + WS_LOGPRI;
    const float* meanReg = ws + WS_MEANREG;
    const int* iws = (const int*)(ws + WS_REGION);

    int eb = blockIdx.x >> 7;   // 128 blocks per expert (512 groups / 4 waves)
    for (int i = threadIdx.x; i < CNUM; i += blockDim.x)
        s_lp[i] = lp[eb * CNUM + i];
    __syncthreads();

    int wid  = threadIdx.x >> 5;
    int lane = threadIdx.x & 31;
    int g  = blockIdx.x * 4 + wid;      // 1536 row-groups total
    int e  = g >> 9;                    // 512 groups per expert
    int b0 = (g & 511) << 4;

    // f32 16x4 A-fragment layout: lane L -> row m=L&15, K pair kb=2*(L>=16)
    int m  = lane & 15;
    int kb = (lane >> 4) << 1;
    const float* gtile0 = logits + ((size_t)e * BNUM + b0) * CNUM; // row b0, col 0
    const float* rowp   = gtile0 + (size_t)m * CNUM;

    v8f acc = {};
    v2f ones; ones[0] = 1.0f; ones[1] = 1.0f;

#if USE_TDM
    {
        unsigned tOff0 = lds_off(&s_tile[wid][0][0]);
        unsigned tOff1 = lds_off(&s_tile[wid][1][0]);
        tdm_issue_tile(gtile0, tOff0);                    // prime tile 0
        for (int t = 0; t < NTILE; ++t) {
            if (t + 1 < NTILE) {
                tdm_issue_tile(gtile0 + (t + 1) * TILE_W,
                               ((t + 1) & 1) ? tOff1 : tOff0);
                __builtin_amdgcn_s_wait_tensorcnt(1);     // tile t landed
            } else {
                __builtin_amdgcn_s_wait_tensorcnt(0);
            }
            __asm__ __volatile__("" ::: "memory");
            const float* tp = &s_tile[wid][t & 1][m * TILE_W];
#pragma unroll
            for (int kc = 0; kc < TILE_W / 4; ++kc) {
                int cl = kc * 4 + kb;
                float2 x = *(const float2*)(tp + cl);     // ds_load_b64
                int gc = t * TILE_W + cl;
                v2f a;
                a[0] = expf(x.x + s_lp[gc]);
                a[1] = expf(x.y + s_lp[gc + 1]);
                acc = __builtin_amdgcn_wmma_f32_16x16x4_f32(
                    false, a, false, ones, (short)0, acc, false, false);
            }
        }
    }
#else
    for (int c0 = 0; c0 < NTILE * TILE_W; c0 += 4) {
        int col = c0 + kb;
        float2 x = *(const float2*)(rowp + col);
        v2f a;
        a[0] = expf(x.x + s_lp[col]);
        a[1] = expf(x.y + s_lp[col + 1]);
        acc = __builtin_amdgcn_wmma_f32_16x16x4_f32(
            false, a, false, ones, (short)0, acc, false, false);
    }
#endif
    // tail columns 960..999 via direct global loads
    for (int c0 = NTILE * TILE_W; c0 < CNUM; c0 += 4) {
        int col = c0 + kb;
        float2 x = *(const float2*)(rowp + col);
        v2f a;
        a[0] = expf(x.x + s_lp[col]);
        a[1] = expf(x.y + s_lp[col + 1]);
        acc = __builtin_amdgcn_wmma_f32_16x16x4_f32(
            false, a, false, ones, (short)0, acc, false, false);
    }

    // D[m][n] = sum_c exp(adj[m][c]) for every n.
    // rows 0..7 live in acc[r] on lanes 0-15; rows 8..15 in acc[r-8] on 16-31.
    int L0 = iws[0], L1 = iws[1], R0 = iws[2], R1 = iws[3];
    float mr0 = meanReg[e * 2], mr1 = meanReg[e * 2 + 1];

    float total = 0.0f;
#pragma unroll
    for (int r = 0; r < 16; ++r) {
        float se = (r < 8) ? __shfl(acc[r], 0, 32) : __shfl(acc[r - 8], 16, 32);
        int b = b0 + r;
        int t = target[b];
        float lraw = logits[((size_t)e * BNUM + b) * CNUM + t];
        float th = acosf(lraw * (1.0f / SMARG));
        int tn = cls[t];
        float d = 0.0f;
        if (tn > L0 && tn <= R0)      d = th - mr0;
        else if (tn > L1 && tn <= R1) d = th - mr1;
        d = (d < 0.0f) ? 1.0f : 1.0f + d;
        total += d * ((lraw + s_lp[t]) - logf(se));   // delta * logp[b, t]
    }
    if (lane == 0) atomicAdd(out, -total * (1.0f / (float)BNUM));
}

extern "C" void kernel_launch(void* const* d_in, const int* in_sizes, int n_in,
                              void* d_out, int out_size, void* d_ws, size_t ws_size,
                              hipStream_t stream) {
    (void)in_sizes; (void)n_in; (void)out_size; (void)ws_size;
    const float* feats  = (const float*)d_in[0];   // [B, 192]
    const float* logits = (const float*)d_in[1];   // [E, B, C]
    const int*   target = (const int*)d_in[2];     // [B]
    const int*   cls    = (const int*)d_in[3];     // [C]
    float* out = (float*)d_out;                    // [1 + E*C]
    float* ws  = (float*)d_ws;

    zero_kernel<<<16, 256, 0, stream>>>(ws, out);
    region_kernel<<<1, 1024, 0, stream>>>(cls, ws);
    persample_kernel<<<BNUM / 8, 256, 0, stream>>>(feats, logits, target, cls, ws);
    stats_kernel<<<ENUM, 1024, 0, stream>>>(cls, ws, out);
    // E*B/16 = 1536 waves, 4 waves/block -> 384 blocks
    loss_kernel<<<384, 128, 0, stream>>>(logits, target, cls, ws, out);
}